// CRF_30219389894781
// MI455X (gfx1250) — compile-verified
//
#include <hip/hip_runtime.h>
#include <math.h>

typedef float v2f __attribute__((ext_vector_type(2)));
typedef float v4f __attribute__((ext_vector_type(4)));
typedef float v8f __attribute__((ext_vector_type(8)));

#define K_TAGS   48
#define T_LEN    1024
#define B_TOT    512
#define START_TAG 46
#define END_TAG   47
#define MINV     (-100000.0f)

// ---------------------------------------------------------------------------
// Gold score: gold[b] = sum_t trans[tag_t, tag_{t-1}] + feats[b,t,tag_t]
//                        + trans[END, tag_{T-1}]      (tag_{-1} = START)
// One 256-thread block per sentence; pure gather + block reduction.
// ---------------------------------------------------------------------------
__global__ __launch_bounds__(256)
void crf_gold_kernel(const float* __restrict__ feats,
                     const float* __restrict__ trans,
                     const int*   __restrict__ tags,
                     float*       __restrict__ gold)
{
    const int b   = blockIdx.x;
    const int tid = threadIdx.x;
    const int*   tb = tags  + (size_t)b * T_LEN;
    const float* fb = feats + (size_t)b * T_LEN * K_TAGS;

    float s = 0.0f;
    for (int t = tid; t < T_LEN; t += 256) {
        const int cur  = tb[t];
        const int prev = (t == 0) ? START_TAG : tb[t - 1];
        s += trans[cur * K_TAGS + prev] + fb[(size_t)t * K_TAGS + cur];
    }

    __shared__ float red[256];
    red[tid] = s;
    __syncthreads();
    #pragma unroll
    for (int off = 128; off > 0; off >>= 1) {
        if (tid < off) red[tid] += red[tid + off];
        __syncthreads();
    }
    if (tid == 0)
        gold[b] = red[0] + trans[END_TAG * K_TAGS + tb[T_LEN - 1]];
}

// ---------------------------------------------------------------------------
// Forward algorithm, 16 sentences per wave, one wave per block.
// Recurrence step as exact-f32 WMMA GEMM:
//   m      = max_k alpha[k]                      (per sentence)
//   p[k]   = exp(alpha[k] - m)
//   s      = E * p          E = exp(trans), 48x48, A-frags resident in VGPRs
//   alpha' = m + log(s) + feat[t]
// ---------------------------------------------------------------------------
__global__ __launch_bounds__(32)
void crf_forward_kernel(const float* __restrict__ feats,
                        const float* __restrict__ trans,
                        const float* __restrict__ gold,
                        float*       __restrict__ out)
{
    __shared__ float ldsE[K_TAGS * K_TAGS];   // exp(trans), row-major
    __shared__ float ldsP[K_TAGS * 16];       // p[state][sentence]
    __shared__ float ldsTE[K_TAGS];           // trans[END][*]

    const int lane = threadIdx.x;             // 0..31 (wave32)
    const int lo   = lane & 15;               // sentence column / M sub-row
    const int hi   = lane >> 4;               // half-wave select
    const int b0   = blockIdx.x * 16;         // first sentence of this wave

    // ---- Build E = exp(trans) and stash trans[END] row --------------------
    for (int idx = lane; idx < K_TAGS * K_TAGS; idx += 32)
        ldsE[idx] = __expf(trans[idx]);       // exp(MINV) underflows to 0
    for (int idx = lane; idx < K_TAGS; idx += 32)
        ldsTE[idx] = trans[END_TAG * K_TAGS + idx];
    __syncthreads();

    // ---- A-matrix fragments of E (ISA 16x4 f32 A layout) ------------------
    // tile (m,c), component v, lane l -> E[16m + (l&15)][4c + v + 2*(l>>4)]
    v2f Ea[3][12];
    #pragma unroll
    for (int m = 0; m < 3; ++m) {
        #pragma unroll
        for (int c = 0; c < 12; ++c) {
            const int row  = 16 * m + lo;
            const int col0 = 4 * c + 2 * hi;
            Ea[m][c].x = ldsE[row * K_TAGS + col0 + 0];
            Ea[m][c].y = ldsE[row * K_TAGS + col0 + 1];
        }
    }

    // ---- alpha in C/D layout: tile m, vgpr r, lane l ----------------------
    //      state i = 16m + r + 8*(l>>4), sentence n = l&15
    v8f alpha[3];
    #pragma unroll
    for (int m = 0; m < 3; ++m) {
        #pragma unroll
        for (int r = 0; r < 8; ++r) {
            const int i = 16 * m + r + 8 * hi;
            alpha[m][r] = (i == START_TAG) ? 0.0f : MINV;
        }
    }

    // Per-lane feat base: row for sentence (b0+lo), pre-offset by 8*hi states
    const float* fbase = feats + (size_t)(b0 + lo) * T_LEN * K_TAGS + 8 * hi;

    for (int t = 0; t < T_LEN; ++t) {
        // Emission loads: independent of the alpha chain -> overlap with WMMA
        v4f f0[3], f1[3];
        #pragma unroll
        for (int m = 0; m < 3; ++m) {
            const float* fp = fbase + (size_t)t * K_TAGS + 16 * m;
            f0[m] = *reinterpret_cast<const v4f*>(fp);
            f1[m] = *reinterpret_cast<const v4f*>(fp + 4);
        }
        if (t + 8 < T_LEN)
            __builtin_prefetch((const void*)(fbase + (size_t)(t + 8) * K_TAGS), 0, 0);

        // ---- per-sentence max over 48 states ----
        float mx = MINV;
        #pragma unroll
        for (int m = 0; m < 3; ++m)
            #pragma unroll
            for (int r = 0; r < 8; ++r)
                mx = fmaxf(mx, alpha[m][r]);
        mx = fmaxf(mx, __shfl_xor(mx, 16));   // combine the two half-waves

        // ---- p = exp(alpha - mx) scattered to canonical [state][n] --------
        #pragma unroll
        for (int m = 0; m < 3; ++m)
            #pragma unroll
            for (int r = 0; r < 8; ++r)
                ldsP[(16 * m + r + 8 * hi) * 16 + lo] = __expf(alpha[m][r] - mx);
        __syncthreads();   // single-wave WG: S_NOP at HW, compiler fence only

        // ---- gather B fragments (4x16 f32 B layout) -----------------------
        // chunk c, component v, lane l -> p[4c + v + 2*(l>>4)][l&15]
        v2f Bt[12];
        #pragma unroll
        for (int c = 0; c < 12; ++c) {
            Bt[c].x = ldsP[(4 * c + 0 + 2 * hi) * 16 + lo];
            Bt[c].y = ldsP[(4 * c + 1 + 2 * hi) * 16 + lo];
        }

        // ---- S = E * P : 3 independent accumulation chains ----------------
        v8f acc[3] = {v8f{}, v8f{}, v8f{}};
        #pragma unroll
        for (int c = 0; c < 12; ++c) {
            #pragma unroll
            for (int m = 0; m < 3; ++m) {
                acc[m] = __builtin_amdgcn_wmma_f32_16x16x4_f32(
                    /*neg_a=*/false, Ea[m][c],
                    /*neg_b=*/false, Bt[c],
                    /*c_mod=*/(short)0, acc[m],
                    /*reuse_a=*/false, /*reuse_b=*/false);
            }
        }

        // ---- alpha' = mx + log(S) + feat ----------------------------------
        #pragma unroll
        for (int m = 0; m < 3; ++m) {
            #pragma unroll
            for (int r = 0; r < 8; ++r) {
                const float fe = (r < 4) ? f0[m][r] : f1[m][r - 4];
                alpha[m][r] = mx + __logf(acc[m][r]) + fe;
            }
        }
        __syncthreads();   // order ldsP reuse across iterations (compiler fence)
    }

    // ---- log_z = logsumexp_k(alpha[k] + trans[END][k]) --------------------
    float v[3][8];
    float mx = MINV;
    #pragma unroll
    for (int m = 0; m < 3; ++m)
        #pragma unroll
        for (int r = 0; r < 8; ++r) {
            v[m][r] = alpha[m][r] + ldsTE[16 * m + r + 8 * hi];
            mx = fmaxf(mx, v[m][r]);
        }
    mx = fmaxf(mx, __shfl_xor(mx, 16));

    float sum = 0.0f;
    #pragma unroll
    for (int m = 0; m < 3; ++m)
        #pragma unroll
        for (int r = 0; r < 8; ++r)
            sum += __expf(v[m][r] - mx);
    sum += __shfl_xor(sum, 16);

    const float logz = mx + __logf(sum);
    if (hi == 0)
        out[b0 + lo] = logz - gold[b0 + lo];
}

// ---------------------------------------------------------------------------
extern "C" void kernel_launch(void* const* d_in, const int* in_sizes, int n_in,
                              void* d_out, int out_size, void* d_ws, size_t ws_size,
                              hipStream_t stream)
{
    const float* feats = (const float*)d_in[0];   // [B, T, K] f32
    const float* trans = (const float*)d_in[1];   // [K, K]    f32
    const int*   tags  = (const int*)  d_in[2];   // [B, T]    i32
    float*       outp  = (float*)d_out;           // [B]       f32
    float*       gold  = (float*)d_ws;            // [B] scratch

    crf_gold_kernel<<<B_TOT, 256, 0, stream>>>(feats, trans, tags, gold);
    crf_forward_kernel<<<B_TOT / 16, 32, 0, stream>>>(feats, trans, gold, outp);
}